// YOLOv3Loss_53764400611731
// MI455X (gfx1250) — compile-verified
//
#include <hip/hip_runtime.h>
#include <math.h>

#define NB   16
#define NA   3
#define HH   85
#define WW   85
#define HWC  (HH * WW)            // 7225
#define MM   50
#define NC   80
#define CCH  (NA * (5 + NC))      // 255
#define STOT (NB * HWC * NA)      // 346800
#define THREADS 256
#define NBLK ((STOT + THREADS - 1) / THREADS)   // 1355
#define NBLKP (((NBLK + 63) / 64) * 64)         // 1408, padded for WMMA reduce
#define IGN  0.7f

typedef float v2f __attribute__((ext_vector_type(2)));
typedef float v8f __attribute__((ext_vector_type(8)));

__device__ __forceinline__ float sigm(float x) { return 1.0f / (1.0f + expf(-x)); }

__device__ __forceinline__ float bce(float p, float t) {
    return -(t * logf(fmaxf(p, 1e-30f)) + (1.0f - t) * logf(fmaxf(1.0f - p, 1e-30f)));
}

__device__ __forceinline__ float iou_cxcywh(float ax, float ay, float aw, float ah,
                                            float bx, float by, float bw, float bh) {
    float tlx = fmaxf(ax - aw * 0.5f, bx - bw * 0.5f);
    float tly = fmaxf(ay - ah * 0.5f, by - bh * 0.5f);
    float brx = fminf(ax + aw * 0.5f, bx + bw * 0.5f);
    float bry = fminf(ay + ah * 0.5f, by + bh * 0.5f);
    float en  = (tlx < brx && tly < bry) ? 1.0f : 0.0f;
    float it  = (brx - tlx) * (bry - tly) * en;
    return it / (aw * ah + bw * bh - it);
}

// ---- kernel 1: per-batch valid-object count -------------------------------
__global__ void k_numobj(const float* __restrict__ tg, int* __restrict__ num_obj) {
    int t = blockIdx.x * blockDim.x + threadIdx.x;
    if (t >= NB * MM) return;
    int b = t / MM, m = t % MM;
    const float* r = tg + (size_t)(b * MM + m) * 5;
    float s = r[0] + r[1] + r[2] + r[3] + r[4];
    if (s > 0.0f) atomicAdd(&num_obj[b], 1);
}

// ---- kernel 2: max IoU per (b,cell,a) + ignore count ----------------------
__global__ void k_maxiou(const float* __restrict__ out, const float* __restrict__ tg,
                         const float* __restrict__ ma, const int* __restrict__ num_obj,
                         float* __restrict__ maxiou, int* __restrict__ npos) {
    int t = blockIdx.x * blockDim.x + threadIdx.x;
    if (t >= STOT) return;
    int b = t / (HWC * NA);
    int r = t - b * (HWC * NA);
    int cell = r / NA, a = r % NA;
    int y = cell / WW, x = cell % WW;
    const float* base = out + ((size_t)b * CCH + (size_t)a * (5 + NC)) * HWC + cell;
    float px = sigm(base[0])        + (float)x;
    float py = sigm(base[HWC])      + (float)y;
    float pw = expf(base[2 * HWC])  * ma[2 * a];
    float ph = expf(base[3 * HWC])  * ma[2 * a + 1];
    int nobj = num_obj[b];
    float mx = -1.0f;
    for (int m = 0; m < nobj; ++m) {
        const float* g = tg + (size_t)(b * MM + m) * 5;
        mx = fmaxf(mx, iou_cxcywh(px, py, pw, ph,
                                  g[0] * WW, g[1] * HH, g[2] * WW, g[3] * HH));
    }
    maxiou[t] = mx;
    if (mx > IGN) atomicAdd(&npos[b], 1);
}

// ---- kernel 3: sequential target-building scan (1 wave per batch) ---------
__global__ void k_scan(const float* __restrict__ tg, const float* __restrict__ ma,
                       const float* __restrict__ ra, const int* __restrict__ am,
                       const int* __restrict__ num_obj, const float* __restrict__ maxiou,
                       float* __restrict__ iou_tgt, float* __restrict__ box_mask,
                       float* __restrict__ box_scale, float* __restrict__ box_tgt,
                       unsigned* __restrict__ cls_bits) {
    int b = blockIdx.x;
    int lane = threadIdx.x;
    int nobj = num_obj[b];
    int am0 = am[0], am1 = am[1], am2 = am[2];
    for (int m = 0; m < MM; ++m) {
        const float* g = tg + (size_t)(b * MM + m) * 5;
        float gx = g[0] * WW, gy = g[1] * HH, gw = g[2] * WW, gh = g[3] * HH;
        int cls = (int)g[4];
        cls = cls < 0 ? 0 : (cls > NC - 1 ? NC - 1 : cls);
        bool valid = (m < nobj);
        int cx = (int)floorf(gx), cy = (int)floorf(gy);
        int cell = cy * WW + cx;
        cell = cell < 0 ? 0 : (cell > HWC - 1 ? HWC - 1 : cell);
        int x = cell % WW, y = cell / WW;
        // lanes 0..8: IoU of the 9 reference anchors at this cell vs the GT box
        float iour = -1e30f;
        if (lane < 9) {
            float off = (lane % 3 == 0) ? 0.5f : 0.0f;   // ref_xcyc adds 0.5 at r=0,3,6
            iour = iou_cxcywh((float)x + off, (float)y + off,
                              ra[2 * lane], ra[2 * lane + 1], gx, gy, gw, gh);
        }
        float best = -1e30f; int aidx = 0;
        for (int r0 = 0; r0 < 9; ++r0) {              // first-max tie (jnp.argmax)
            float v = __shfl(iour, r0, 32);
            if (v > best) { best = v; aidx = r0; }
        }
        if (lane == 0) {
            bool doo = valid && (aidx == am0 || aidx == am1 || aidx == am2);
            if (doo) {
                int a3 = aidx % 3;
                int s = (b * HWC + cell) * NA + a3;
                iou_tgt[s]   = maxiou[s];
                box_mask[s]  = 1.0f;
                box_scale[s] = sqrtf(2.0f - gw * gh / (float)(WW * HH));
                box_tgt[4 * s + 0] = gx - (float)x;
                box_tgt[4 * s + 1] = gy - (float)y;
                box_tgt[4 * s + 2] = gw / ma[2 * a3];
                box_tgt[4 * s + 3] = gh / ma[2 * a3 + 1];
                cls_bits[3 * s + (cls >> 5)] |= (1u << (cls & 31));
            }
        }
    }
}

// ---- kernel 4: elementwise loss, block partial sums -----------------------
__global__ void k_loss(const float* __restrict__ out, const int* __restrict__ num_obj,
                       const int* __restrict__ npos, const float* __restrict__ maxiou,
                       const float* __restrict__ iou_tgt, const float* __restrict__ box_mask,
                       const float* __restrict__ box_scale, const float* __restrict__ box_tgt,
                       const unsigned* __restrict__ cls_bits, float* __restrict__ partials) {
    __shared__ float sm[THREADS];
    int t = blockIdx.x * THREADS + threadIdx.x;
    float l = 0.0f;
    if (t < STOT) {
        int b = t / (HWC * NA);
        int r = t - b * (HWC * NA);
        int cell = r / NA, a = r % NA;
        const float* base = out + ((size_t)b * CCH + (size_t)a * (5 + NC)) * HWC + cell;
        float tx = sigm(base[0]);
        float ty = sigm(base[HWC]);
        float tw = expf(base[2 * HWC]);
        float th = expf(base[3 * HWC]);
        float conf = sigm(base[4 * HWC]);
        float bm = box_mask[t];
        if (bm > 0.0f) {                      // matched cell: box + class terms
            __builtin_prefetch(base + 5 * HWC, 0, 1);   // global_prefetch_b8
            float sc = box_scale[t];
            float w2 = sc * sc;
            float b0 = box_tgt[4 * t + 0], b1 = box_tgt[4 * t + 1];
            float b2 = box_tgt[4 * t + 2], b3 = box_tgt[4 * t + 3];
            l += w2 * (bce(tx, b0) + bce(ty, b1));
            float dw = (tw - b2) * sc, dh = (th - b3) * sc;
            l += 0.5f * (dw * dw + dh * dh);
            unsigned c0 = cls_bits[3 * t], c1 = cls_bits[3 * t + 1], c2 = cls_bits[3 * t + 2];
            for (int c = 0; c < NC; ++c) {
                float p = sigm(base[(5 + c) * HWC]);
                unsigned wbits = (c < 32) ? c0 : ((c < 64) ? c1 : c2);
                float tt = (float)((wbits >> (c & 31)) & 1u);
                l += bce(p, tt);
            }
        }
        // objectness term for every cell (unmatched masked cells contribute 0)
        float im;
        int nb = num_obj[b];
        if (nb == 0)            im = 0.0f;
        else if (bm > 0.0f)     im = 1.0f;
        else                    im = (npos[b] > 0 && maxiou[t] >= IGN) ? 0.0f : 1.0f;
        l += bce(conf * im, iou_tgt[t] * im);
    }
    sm[threadIdx.x] = l;
    __syncthreads();
    for (int st = THREADS / 2; st > 0; st >>= 1) {
        if (threadIdx.x < st) sm[threadIdx.x] += sm[threadIdx.x + st];
        __syncthreads();
    }
    if (threadIdx.x == 0) partials[blockIdx.x] = sm[0];
}

// ---- kernel 5: exact final reduction via V_WMMA_F32_16X16X4_F32 -----------
// A = ones(16x4)  =>  D[m][n] = sum_k B[k][n] + C[m][n].  We feed 64 partials
// per iteration into B's 64 element slots via a bijective lane->element map
// (lane loads partials[base+2*lane .. +1] as one b64), so sum_n D[0][n] is the
// exact f32 running total regardless of the hardware's (k,n) assignment.
// Tail beyond NBLK is zero-padded workspace, so no bounds checks: the loop is
// a straight global_load_b64 + v_wmma with EXEC all ones throughout.
__global__ void k_reduce(const float* __restrict__ partials, float* __restrict__ outp) {
    int lane = threadIdx.x;           // single wave32, EXEC all ones
    v8f acc = {};
    v2f ones; ones.x = 1.0f; ones.y = 1.0f;
    for (int base = 0; base < NBLKP; base += 64) {
        v2f bv = *(const v2f*)(partials + base + 2 * lane);
        acc = __builtin_amdgcn_wmma_f32_16x16x4_f32(
            /*neg_a=*/false, ones, /*neg_b=*/false, bv,
            /*c_mod=*/(short)0, acc, /*reuse_a=*/false, /*reuse_b=*/false);
    }
    float v = (lane < 16) ? acc[0] : 0.0f;    // D row 0 lives in VGPR0, lanes 0..15
    for (int off = 16; off > 0; off >>= 1) v += __shfl_xor(v, off, 32);
    if (lane == 0) outp[0] = v;
}

extern "C" void kernel_launch(void* const* d_in, const int* in_sizes, int n_in,
                              void* d_out, int out_size, void* d_ws, size_t ws_size,
                              hipStream_t stream) {
    const float* outputs = (const float*)d_in[0];
    const float* targets = (const float*)d_in[1];
    const float* ma      = (const float*)d_in[2];
    const float* ra      = (const float*)d_in[3];
    const int*   am      = (const int*)d_in[4];

    char* ws = (char*)d_ws;
    size_t off = 0;
    auto alloc = [&](size_t bytes) -> char* {
        char* p = ws + off;
        off += (bytes + 255) & ~(size_t)255;
        return p;
    };
    // no-zero-needed region (fully overwritten every call before being read)
    float*    maxiou    = (float*)alloc((size_t)STOT * 4);
    float*    box_scale = (float*)alloc((size_t)STOT * 4);
    float*    box_tgt   = (float*)alloc((size_t)STOT * 16);
    // contiguous zero-init region (defaults must be 0; partials tail must be 0)
    char* zstart = ws + off;
    float*    iou_tgt   = (float*)alloc((size_t)STOT * 4);
    float*    box_mask  = (float*)alloc((size_t)STOT * 4);
    unsigned* cls_bits  = (unsigned*)alloc((size_t)STOT * 12);
    float*    partials  = (float*)alloc((size_t)NBLKP * 4);
    int*      npos      = (int*)alloc((size_t)NB * 4);
    int*      num_obj   = (int*)alloc((size_t)NB * 4);
    size_t zbytes = (size_t)((ws + off) - zstart);

    hipMemsetAsync(zstart, 0, zbytes, stream);
    k_numobj<<<(NB * MM + 255) / 256, 256, 0, stream>>>(targets, num_obj);
    k_maxiou<<<NBLK, THREADS, 0, stream>>>(outputs, targets, ma, num_obj, maxiou, npos);
    k_scan<<<NB, 32, 0, stream>>>(targets, ma, ra, am, num_obj, maxiou,
                                  iou_tgt, box_mask, box_scale, box_tgt, cls_bits);
    k_loss<<<NBLK, THREADS, 0, stream>>>(outputs, num_obj, npos, maxiou, iou_tgt,
                                         box_mask, box_scale, box_tgt, cls_bits, partials);
    k_reduce<<<1, 32, 0, stream>>>(partials, (float*)d_out);
}